// OcrEmbedding_12206297055340
// MI455X (gfx1250) — compile-verified
//
#include <hip/hip_runtime.h>
#include <hip/hip_bf16.h>

// OcrEmbedding: EmbeddingBag-sum gather.
//   ids   : int32  [B=32, T=512, S=4]   (d_in[0])
//   table : float  [V=50000, D=256]     (d_in[1])
//   out   : float  [B, T, D]            (d_out)
// out[b,t,:] = sum_s ( ids[b,t,s] != 0 ? table[ids[b,t,s], :] : 0 )
//
// Pure bandwidth kernel. Table (51.2 MB) fits in MI455X's 192 MB L2, so
// gathers should stay L2-resident; output is streamed with non-temporal
// b128 stores to avoid evicting the table.

typedef float v4f __attribute__((ext_vector_type(4)));
typedef int   v4i __attribute__((ext_vector_type(4)));

#define EMB_D      256
#define EMB_S      4
#define CHUNKS     (EMB_D / 4)   // 64 float4 chunks per output row

__global__ __launch_bounds__(256)
void OcrEmbedding_bagsum_kernel(const int*   __restrict__ ids,
                                const float* __restrict__ table,
                                float*       __restrict__ out,
                                int n_rows)   // n_rows = B*T
{
    const int tid = blockIdx.x * blockDim.x + threadIdx.x;
    if (tid >= n_rows * CHUNKS) return;   // grid is exact; kept for safety

    // 64 consecutive tids share one token row; a wave32 (32-aligned run of
    // 32 tids) always lies inside one such group, so the row id is
    // wave-uniform. readfirstlane makes that explicit -> scalar index load.
    const int row   = __builtin_amdgcn_readfirstlane(tid >> 6);
    const int chunk = tid & (CHUNKS - 1);

    // One s_load_b128 per wave for all four subtoken indices.
    const v4i idx = *reinterpret_cast<const v4i*>(ids + row * EMB_S);

    const v4f* __restrict__ t4 = reinterpret_cast<const v4f*>(table);

    // Four b128 gathers (issued back-to-back; LOADcnt hides latency).
    const v4f g0 = t4[idx.x * CHUNKS + chunk];
    const v4f g1 = t4[idx.y * CHUNKS + chunk];
    const v4f g2 = t4[idx.z * CHUNKS + chunk];
    const v4f g3 = t4[idx.w * CHUNKS + chunk];

    // padding_idx==0 rows must contribute zero (the provided table row 0 is
    // NOT zero). Branchless mask keeps EXEC full: 16 v_fma total.
    const float m0 = (idx.x != 0) ? 1.0f : 0.0f;
    const float m1 = (idx.y != 0) ? 1.0f : 0.0f;
    const float m2 = (idx.z != 0) ? 1.0f : 0.0f;
    const float m3 = (idx.w != 0) ? 1.0f : 0.0f;

    v4f acc = g0 * m0;
    acc += g1 * m1;
    acc += g2 * m2;
    acc += g3 * m3;

    // Streaming write-once output: non-temporal b128 store so the 16 MB
    // output does not evict the L2-resident embedding table.
    __builtin_nontemporal_store(acc, reinterpret_cast<v4f*>(out) + tid);
}

extern "C" void kernel_launch(void* const* d_in, const int* in_sizes, int n_in,
                              void* d_out, int out_size, void* d_ws, size_t ws_size,
                              hipStream_t stream)
{
    const int*   ids   = reinterpret_cast<const int*>(d_in[0]);
    const float* table = reinterpret_cast<const float*>(d_in[1]);
    float*       out   = reinterpret_cast<float*>(d_out);

    const int n_rows  = in_sizes[0] / EMB_S;        // B*T = 16384
    const int threads = n_rows * CHUNKS;            // 1,048,576
    const int block   = 256;                        // 8 wave32 per block
    const int grid    = (threads + block - 1) / block;

    OcrEmbedding_bagsum_kernel<<<grid, block, 0, stream>>>(ids, table, out, n_rows);
}